// NeuralNetwork_17360257811096
// MI455X (gfx1250) — compile-verified
//
#include <hip/hip_runtime.h>

// CDNA5 / gfx1250: f32 WMMA (V_WMMA_F32_16X16X4_F32) batched MLP, wave32.
// D = A(16x4 f32) * B(4x16 f32) + C(16x16 f32). One 16-row stripe per wave-tile.

typedef float v2f __attribute__((ext_vector_type(2)));
typedef float v4f __attribute__((ext_vector_type(4)));
typedef float v8f __attribute__((ext_vector_type(8)));

#define NBOARDS        262144
#define ROWS           (NBOARDS * 5)    // 1,310,720 rows of 5
#define WAVES_PER_WG   4
#define TILES_PER_WAVE 10
#define SA 124                          // LDS stride, 120-wide buffer (+4 pad; rows 16B aligned)
#define SB 100                          // LDS stride, 96-wide buffer (+4 pad)
#define WAVE_LDS (16 * SA + 16 * SB + 80 + 80)  // bufA + bufB + x-stage + out-stage = 3744 floats
#define W1TAB    1024                   // padded B1 (8x128) in (k-pair, n, parity) layout
#define B1TAB    128                    // padded bias1 (120 -> 128)

static __device__ __forceinline__ v8f wmma_f32(v2f a, v2f b, v8f c) {
  // (neg_a, A, neg_b, B, c_mod, C, reuse_a, reuse_b)
  return __builtin_amdgcn_wmma_f32_16x16x4_f32(false, a, false, b, (short)0, c,
                                               false, false);
}

// ReLU as a single v_med3_f32 (clamp to [0, +inf)) -- avoids the canonicalize+max
// pair that fmaxf(x, 0.f) lowers to.
static __device__ __forceinline__ float relu1(float v) {
  return __builtin_amdgcn_fmed3f(v, 0.f, __builtin_inff());
}

// 128 threads/block; require >= 4 waves per SIMD => <= 256 VGPRs/wave.
__global__ __launch_bounds__(WAVES_PER_WG * 32, 4) void
mlp5_wmma_kernel(const float* __restrict__ x,
                 const float* __restrict__ W1, const float* __restrict__ b1,
                 const float* __restrict__ W2, const float* __restrict__ b2,
                 const float* __restrict__ W3, const float* __restrict__ b3,
                 float* __restrict__ out)
{
  __shared__ float smem[WAVES_PER_WG * WAVE_LDS + W1TAB + B1TAB];   // 64,512 bytes

  const int lane = threadIdx.x & 31;
  const int wave = threadIdx.x >> 5;
  float* bufA = smem + wave * WAVE_LDS;   // 16 x 120(+4) y1 (reused for y3 tile)
  float* bufB = bufA + 16 * SA;           // 16 x 96(+4)  y2
  float* xs   = bufB + 16 * SB;           // 80 floats: this tile's x rows (contiguous)
  float* os   = xs + 80;                  // 80 floats: normalized outputs (pre-mask)
  float* w1t  = smem + WAVES_PER_WG * WAVE_LDS;  // block-shared B1 fragment table
  float* b1t  = w1t + W1TAB;                     // block-shared padded bias1

  // Fragment index helpers (ISA 7.12.2, wave32):
  const int m    = lane & 15;             // A-matrix row (M)
  const int kb   = (lane >> 4) << 1;      // K sub-offset in a 4-chunk: 0 or 2
  const int ncol = lane & 15;             // B/C/D column within 16-wide tile
  const int mrow = (lane >> 4) << 3;      // C/D: VGPR j holds row mrow + j

  // ---- Fill B1 + bias1 tables once per block.
  // B1[k][n] = W1[n][k], 5x120 zero-padded to 8x128, laid out so a (k0,k0+1)
  // fragment is one aligned ds_load_b64: w1t[(k>>1)*256 + n*2 + (k&1)].
  for (int i = threadIdx.x; i < W1TAB; i += WAVES_PER_WG * 32) {
    const int kp  = i >> 8;        // k-pair 0..3
    const int n   = (i & 255) >> 1;
    const int par = i & 1;
    const int k   = kp * 2 + par;
    w1t[i] = (k < 5 && n < 120) ? W1[n * 5 + k] : 0.f;
  }
  if (threadIdx.x < B1TAB)
    b1t[threadIdx.x] = (threadIdx.x < 120) ? b1[threadIdx.x] : 0.f;
  __syncthreads();

  const int tile0 = (blockIdx.x * WAVES_PER_WG + wave) * TILES_PER_WAVE;

  for (int t = 0; t < TILES_PER_WAVE; ++t) {
    const int rowbase = (tile0 + t) * 16;
    const size_t gbase = (size_t)rowbase * 5;      // 80 contiguous floats per tile

    // ---- Stage x tile: 20 lanes x b128 coalesced global load -> LDS ----
    if (lane < 20) {
      v4f xv = *(const v4f*)(x + gbase + lane * 4);
      *(v4f*)(xs + lane * 4) = xv;
    }

    // ---- Layer-1 A fragments from LDS stage ----
    v2f a0 = {xs[m * 5 + kb], xs[m * 5 + kb + 1]};       // K = kb, kb+1 (< 5 always)
    float x4 = xs[m * 5 + 4];
    v2f a1 = {(kb == 0) ? x4 : 0.f, 0.f};                // K = 4 real, 5..7 zero pad

    // ---- Layer 1: y1[16x120] = relu(X[16x8] * B1[8x128] + b1) ----
    const int kpb = kb >> 1;                             // k-pair sub-offset: 0 or 1
#pragma unroll
    for (int nt = 0; nt < 8; ++nt) {
      const int n = nt * 16 + ncol;
      const float bias = b1t[n];                         // padded: no clamp needed
      v2f bf0 = *(const v2f*)(w1t + (0 * 2 + kpb) * 256 + n * 2);  // kc=0
      v2f bf1 = *(const v2f*)(w1t + (1 * 2 + kpb) * 256 + n * 2);  // kc=1
      v8f c = {bias, bias, bias, bias, bias, bias, bias, bias};
      c = wmma_f32(a0, bf0, c);
      c = wmma_f32(a1, bf1, c);
      if (nt < 7 || ncol < 8) {            // bufA holds only cols 0..119 (+pad)
#pragma unroll
        for (int j = 0; j < 8; ++j)
          bufA[(mrow + j) * SA + nt * 16 + ncol] = relu1(c[j]);
      }
    }

    // ---- Layer 2: y2 = relu(y1[16x120] * B2[120x96] + b2); B2[k][n] = W2[n][k] ----
    // A fragments re-read from LDS per n-tile (ds_load_b64, 8B aligned) to keep
    // VGPR pressure < 256; n-tiles 0..4 are always in-bounds (n <= 79 < 84).
    const float* aAp = bufA + m * SA + kb;     // A-frag base for layer 2
#pragma unroll 1
    for (int nt = 0; nt < 5; ++nt) {
      const int n = nt * 16 + ncol;
      const float* w2p = W2 + (size_t)n * 120;
      const float bias = b2[n];
      v8f c0 = {bias, bias, bias, bias, bias, bias, bias, bias};
      v8f c1 = {0.f, 0.f, 0.f, 0.f, 0.f, 0.f, 0.f, 0.f};
#pragma unroll
      for (int kc = 0; kc < 30; kc += 2) {     // dual accumulators: 15+15 chains
        v2f aA = *(const v2f*)(aAp + kc * 4);
        v2f aB = *(const v2f*)(aAp + (kc + 1) * 4);
        v2f bA = *(const v2f*)(w2p + kc * 4 + kb);
        v2f bB = *(const v2f*)(w2p + (kc + 1) * 4 + kb);
        c0 = wmma_f32(aA, bA, c0);
        c1 = wmma_f32(aB, bB, c1);
      }
      v8f c = c0 + c1;
#pragma unroll
      for (int j = 0; j < 8; ++j)
        bufB[(mrow + j) * SB + nt * 16 + ncol] = relu1(c[j]);
    }
    // n-tile 5 (n = 80..95): columns >= 84 are zero-padded.
    {
      const int n = 80 + ncol;
      const float* w2p = W2 + (size_t)(n < 84 ? n : 83) * 120;
      const float nm = (n < 84) ? 1.f : 0.f;
      float bias = b2[(n < 84) ? n : 83];
      bias = (n < 84) ? bias : 0.f;
      v8f c0 = {bias, bias, bias, bias, bias, bias, bias, bias};
      v8f c1 = {0.f, 0.f, 0.f, 0.f, 0.f, 0.f, 0.f, 0.f};
#pragma unroll
      for (int kc = 0; kc < 30; kc += 2) {
        v2f aA = *(const v2f*)(aAp + kc * 4);
        v2f aB = *(const v2f*)(aAp + (kc + 1) * 4);
        v2f bA = *(const v2f*)(w2p + kc * 4 + kb);
        v2f bB = *(const v2f*)(w2p + (kc + 1) * 4 + kb);
        c0 = wmma_f32(aA, bA * nm, c0);
        c1 = wmma_f32(aB, bB * nm, c1);
      }
      v8f c = c0 + c1;
#pragma unroll
      for (int j = 0; j < 8; ++j)
        bufB[(mrow + j) * SB + 80 + ncol] = relu1(c[j]);
    }

    // ---- Layer 3: y3[16x16] = relu(y2[16x84] * B3[84x16] + b3); only n<5 real ----
    {
      const float* aBp = bufB + m * SB + kb;
      const float* w3p = W3 + (size_t)(ncol < 5 ? ncol : 4) * 84;
      const float nm = (ncol < 5) ? 1.f : 0.f;
      float bias = b3[(ncol < 5) ? ncol : 4];
      bias = (ncol < 5) ? bias : 0.f;
      v8f c0 = {bias, bias, bias, bias, bias, bias, bias, bias};
      v8f c1 = {0.f, 0.f, 0.f, 0.f, 0.f, 0.f, 0.f, 0.f};
#pragma unroll
      for (int kc = 0; kc < 20; kc += 2) {     // 11 + 10 chains
        v2f aA = *(const v2f*)(aBp + kc * 4);
        v2f aB = *(const v2f*)(aBp + (kc + 1) * 4);
        v2f bA = *(const v2f*)(w3p + kc * 4 + kb);
        v2f bB = *(const v2f*)(w3p + (kc + 1) * 4 + kb);
        c0 = wmma_f32(aA, bA * nm, c0);
        c1 = wmma_f32(aB, bB * nm, c1);
      }
      {
        v2f aA = *(const v2f*)(aBp + 20 * 4);
        v2f bA = *(const v2f*)(w3p + 20 * 4 + kb);
        c0 = wmma_f32(aA, bA * nm, c0);
      }
      v8f c = c0 + c1;
#pragma unroll
      for (int j = 0; j < 8; ++j)
        bufA[(mrow + j) * SA + ncol] = relu1(c[j]);   // y3 tile -> bufA cols 0..15
    }

    // ---- Epilogue phase 1: per-row L2 norm (5 elems), normalized -> LDS stage ----
    {
      const int r16 = lane >> 1;                 // 2 lanes per row
      v4f y4v = *(const v4f*)(bufA + r16 * SA);  // cols 0..3 (16B aligned: SA*4=496)
      const float y4e = bufA[r16 * SA + 4];      // col 4
      const float ss = y4v.x * y4v.x + y4v.y * y4v.y + y4v.z * y4v.z +
                       y4v.w * y4v.w + y4e * y4e;
      const float inv = 1.f / fmaxf(sqrtf(ss), 1e-12f);
      if ((lane & 1) == 0) {
        os[r16 * 5 + 0] = y4v.x * inv;
        os[r16 * 5 + 1] = y4v.y * inv;
        os[r16 * 5 + 2] = y4v.z * inv;
      } else {
        os[r16 * 5 + 3] = y4v.w * inv;
        os[r16 * 5 + 4] = y4e * inv;
      }
    }

    // ---- Epilogue phase 2: mask with x!=0 (from LDS x stage), coalesced b128 out ----
    if (lane < 20) {
      v4f yv = *(const v4f*)(os + lane * 4);
      v4f xv = *(const v4f*)(xs + lane * 4);
      v4f o;
      o.x = (xv.x != 0.f) ? 0.f : yv.x;
      o.y = (xv.y != 0.f) ? 0.f : yv.y;
      o.z = (xv.z != 0.f) ? 0.f : yv.z;
      o.w = (xv.w != 0.f) ? 0.f : yv.w;
      *(v4f*)(out + gbase + lane * 4) = o;
    }
  }
}

extern "C" void kernel_launch(void* const* d_in, const int* in_sizes, int n_in,
                              void* d_out, int out_size, void* d_ws, size_t ws_size,
                              hipStream_t stream) {
  const float* x  = (const float*)d_in[0];
  const float* W1 = (const float*)d_in[1];
  const float* b1 = (const float*)d_in[2];
  const float* W2 = (const float*)d_in[3];
  const float* b2 = (const float*)d_in[4];
  const float* W3 = (const float*)d_in[5];
  const float* b3 = (const float*)d_in[6];
  float* out = (float*)d_out;

  // ROWS / (16 rows * 10 tiles * 4 waves) = 1,310,720 / 640 = 2048 blocks, exact.
  dim3 grid(ROWS / (16 * TILES_PER_WAVE * WAVES_PER_WG));
  dim3 block(WAVES_PER_WG * 32);
  hipLaunchKernelGGL(mlp5_wmma_kernel, grid, block, 0, stream,
                     x, W1, b1, W2, b2, W3, b3, out);
}